// Tab2DQuantileEmbeddingX_89988154786516
// MI455X (gfx1250) — compile-verified
//
#include <hip/hip_runtime.h>
#include <math.h>

typedef __attribute__((ext_vector_type(2))) float v2f;
typedef __attribute__((ext_vector_type(8))) float v8f;

#define B_    8
#define NS_   1024
#define NQ_   1024
#define NF_   32
#define NQUANT 999
#define PADV  9999.0f

// ---------------------------------------------------------------------------
// WMMA-based sum of 1024 floats residing in LDS. Called by a full wave
// (lanes 0..31, EXEC all ones). B matrix = all ones, so D[m][n] = row-sum(m)
// replicated across n; accumulate across 16 tiles of 64 values each, then
// fold rows: lanes 0-15 hold M=0..7 in c[0..7], lanes 16-31 hold M=8..15.
// ---------------------------------------------------------------------------
__device__ __forceinline__ float wmma_sum_1024(const float* red, int lane) {
    v2f ones; ones.x = 1.0f; ones.y = 1.0f;
    v8f c = {};
#pragma unroll
    for (int t = 0; t < 16; ++t) {
        v2f a;
        a.x = red[t * 64 + lane];
        a.y = red[t * 64 + 32 + lane];
        // 8 args: (neg_a, A, neg_b, B, c_mod, C, reuse_a, reuse_b)
        c = __builtin_amdgcn_wmma_f32_16x16x4_f32(
                false, a, false, ones, (short)0, c, false, false);
    }
    float part = c[0] + c[1] + c[2] + c[3] + c[4] + c[5] + c[6] + c[7];
    part += __shfl_xor(part, 16, 32);   // add M=0..7 block to M=8..15 block
    return part;                        // total on every lane
}

// ---------------------------------------------------------------------------
// Kernel 1: seq_len[b] = count of non-padded support rows (as float)
// ---------------------------------------------------------------------------
__global__ __launch_bounds__(256)
void seqlen_kernel(const unsigned char* __restrict__ mask,
                   float* __restrict__ seq) {
    __shared__ float red[256];
    const int b = blockIdx.x, tid = threadIdx.x;
    float c = 0.0f;
#pragma unroll
    for (int r = 0; r < 4; ++r)
        c += (mask[b * NS_ + tid + r * 256] == 0) ? 1.0f : 0.0f;
    red[tid] = c;
    __syncthreads();
    for (int s = 128; s > 0; s >>= 1) {
        if (tid < s) red[tid] += red[tid + s];
        __syncthreads();
    }
    if (tid == 0) seq[b] = red[0];
}

// ---------------------------------------------------------------------------
// Kernel 2: per (b,f) column: bitonic sort 1024 values in LDS (padded -> 9999
// sorts to the end), then write 999 interpolated quantiles to workspace.
// ---------------------------------------------------------------------------
__global__ __launch_bounds__(256)
void sort_quant_kernel(const float* __restrict__ xs,
                       const unsigned char* __restrict__ mask,
                       float* __restrict__ qws) {
    __shared__ float sbuf[NS_];
    const int col = blockIdx.x;          // b*32 + f
    const int b = col >> 5, f = col & 31;
    const int tid = threadIdx.x;

#pragma unroll
    for (int r = 0; r < 4; ++r) {
        const int s = tid + r * 256;
        const unsigned char m = mask[b * NS_ + s];
        sbuf[s] = m ? PADV : xs[((size_t)b * NS_ + s) * NF_ + f];
    }

    // bitonic sort, ascending
    for (int k = 2; k <= NS_; k <<= 1) {
        for (int j = k >> 1; j > 0; j >>= 1) {
            __syncthreads();
#pragma unroll
            for (int r = 0; r < 4; ++r) {
                const int i = tid + r * 256;
                const int ixj = i ^ j;
                if (ixj > i) {
                    const float a = sbuf[i];
                    const float c = sbuf[ixj];
                    if ((a > c) == ((i & k) == 0)) { sbuf[i] = c; sbuf[ixj] = a; }
                }
            }
        }
    }
    __syncthreads();

    float* qcol = qws + (size_t)col * 1024;
    for (int kq = tid; kq < NQUANT; kq += 256) {
        const float qv   = (float)(kq + 1) / 1000.0f;     // match ref ops
        const float idxf = qv * (float)(NS_ - 1);
        int lo = (int)idxf;
        int hi = lo + 1; if (hi > NS_ - 1) hi = NS_ - 1;
        const float frac = idxf - (float)lo;
        const float lov = sbuf[lo], hiv = sbuf[hi];
        qcol[kq] = lov + frac * (hiv - lov);
    }
}

// ---------------------------------------------------------------------------
// Kernel 3: per (b,f) column: binary-search bucketize support+query, WMMA
// reductions for mean/var, normalize, write both outputs.
// ---------------------------------------------------------------------------
__global__ __launch_bounds__(256)
void embed_kernel(const float* __restrict__ xs,
                  const float* __restrict__ xq,
                  const unsigned char* __restrict__ mask,
                  const float* __restrict__ qws,
                  const float* __restrict__ seq,
                  float* __restrict__ out) {
    __shared__ float qb[1024];
    __shared__ float red[NS_];
    __shared__ float bc[2];

    const int col = blockIdx.x;
    const int b = col >> 5, f = col & 31;
    const int tid = threadIdx.x;

    const float* qcol = qws + (size_t)col * 1024;
#pragma unroll
    for (int r = 0; r < 4; ++r)
        qb[tid + r * 256] = qcol[tid + r * 256];   // indices >=999 never probed
    const float seqL = seq[b];
    __syncthreads();

    // ---- support bucketize -------------------------------------------------
    float vals[4]; bool valid[4];
#pragma unroll
    for (int r = 0; r < 4; ++r) {
        const int s = tid + r * 256;
        const bool ok = (mask[b * NS_ + s] == 0);
        float v = 0.0f;
        if (ok) {
            const float x = xs[((size_t)b * NS_ + s) * NF_ + f];
            int lo = 0, hi = NQUANT;                 // upper bound: mid <= 998
            while (lo < hi) {
                const int mid = (lo + hi) >> 1;
                if (qb[mid] <= x) lo = mid + 1; else hi = mid;
            }
            v = (float)lo / seqL;
        }
        vals[r] = v; valid[r] = ok;
        red[s] = v;                                  // masked entries are 0
    }
    __syncthreads();

    // ---- sum via WMMA ------------------------------------------------------
    if (tid < 32) {
        const float S = wmma_sum_1024(red, tid);
        if (tid == 0) bc[0] = S;
    }
    __syncthreads();
    const float mean = bc[0] / seqL;

#pragma unroll
    for (int r = 0; r < 4; ++r) {
        const int s = tid + r * 256;
        const float d = vals[r] - mean;
        red[s] = valid[r] ? d * d : 0.0f;
    }
    __syncthreads();

    if (tid < 32) {
        const float Q = wmma_sum_1024(red, tid);
        if (tid == 0) bc[1] = Q;
    }
    __syncthreads();

    const float var  = bc[1] / seqL;
    const bool  zv   = (var == 0.0f);
    const float stdv = sqrtf(zv ? 1.0f : var);

    // ---- support outputs ---------------------------------------------------
#pragma unroll
    for (int r = 0; r < 4; ++r) {
        const int s = tid + r * 256;
        const float xc = valid[r] ? (vals[r] - mean) : 0.0f;
        out[((size_t)b * NS_ + s) * NF_ + f] = zv ? 0.0f : xc / stdv;
    }

    // ---- query bucketize + outputs ----------------------------------------
    float* outq = out + (size_t)B_ * NS_ * NF_;
#pragma unroll
    for (int r = 0; r < 4; ++r) {
        const int qi = tid + r * 256;
        const float x = xq[((size_t)b * NQ_ + qi) * NF_ + f];
        int lo = 0, hi = NQUANT;
        while (lo < hi) {
            const int mid = (lo + hi) >> 1;
            if (qb[mid] <= x) lo = mid + 1; else hi = mid;
        }
        const float xqb = (float)lo / seqL;
        outq[((size_t)b * NQ_ + qi) * NF_ + f] = zv ? 0.0f : (xqb - mean) / stdv;
    }
}

// ---------------------------------------------------------------------------
extern "C" void kernel_launch(void* const* d_in, const int* in_sizes, int n_in,
                              void* d_out, int out_size, void* d_ws, size_t ws_size,
                              hipStream_t stream) {
    const float*         xs   = (const float*)d_in[0];          // (8,1024,32) f32
    const float*         xq   = (const float*)d_in[1];          // (8,1024,32) f32
    const unsigned char* mask = (const unsigned char*)d_in[2];  // (8,1024) bool (1B)
    // d_in[3] feature_mask: unused by reference

    float* out = (float*)d_out;          // xs_out ++ xq_out, 524288 floats
    float* seq = (float*)d_ws;           // 8 floats
    float* qws = (float*)d_ws + 16;      // 256 cols * 1024 floats (~1 MB)

    seqlen_kernel    <<<B_,        256, 0, stream>>>(mask, seq);
    sort_quant_kernel<<<B_ * NF_,  256, 0, stream>>>(xs, mask, qws);
    embed_kernel     <<<B_ * NF_,  256, 0, stream>>>(xs, xq, mask, qws, seq, out);
}